// HetConv2d_87935160418930
// MI455X (gfx1250) — compile-verified
//
#include <hip/hip_runtime.h>
#include <stdint.h>

typedef __attribute__((ext_vector_type(2))) float v2f;
typedef __attribute__((ext_vector_type(8))) float v8f;

// ---- geometry ----
#define BATCH 16
#define IC    128
#define OC    128
#define HH    128
#define WW    128
#define NPIX  32          // pixels (N) per block
#define NT    (NPIX/16)   // N-tiles per wave

static __device__ __forceinline__ v8f wmma_f32_16x16x4(v2f a, v2f b, v8f c) {
  // 8 args: (neg_a, A, neg_b, B, c_mod, C, reuse_a, reuse_b)
  return __builtin_amdgcn_wmma_f32_16x16x4_f32(false, a, false, b, (short)0, c,
                                               false, false);
}

// Async DMA one fp32 element from global memory into LDS (per active lane).
// GV mode: 64-bit global address in VGPR pair, LDS byte address in VGPR.
// (LDS byte address == low 32 bits of the generic LDS pointer, per the
//  gfx1250 aperture rules.)
static __device__ __forceinline__ void async_load_f32(void* lds_ptr,
                                                      const float* gptr) {
  uint32_t lds_addr = (uint32_t)(uintptr_t)lds_ptr;
  asm volatile("global_load_async_to_lds_b32 %0, %1, off"
               :
               : "v"(lds_addr), "v"(gptr)
               : "memory");
}

static __device__ __forceinline__ void wait_asynccnt0() {
#if defined(__has_builtin) && __has_builtin(__builtin_amdgcn_s_wait_asynccnt)
  __builtin_amdgcn_s_wait_asynccnt(0);
#else
  asm volatile("s_wait_asynccnt 0x0" ::: "memory");
#endif
}

// Residue-major channel permutation: c' = (c%4)*32 + c/4, both for output
// channels (o') and input channels (i').  Under this order, residue group r
// occupies the contiguous slot range [32r, 32r+32).
//
//  WcR[o'][i']               : dense center weights (3x3 center tap or 1x1)
//  WgR[r][tap(8)][og][ig]    : grouped off-center 3x3 taps (residue groups)
//  biasR[o']                 : sum_i b[o][i]
__global__ void prep_weights(const float* __restrict__ w3,
                             const float* __restrict__ w1,
                             const float* __restrict__ bsrc,
                             float* __restrict__ WcR,
                             float* __restrict__ WgR,
                             float* __restrict__ biasR) {
  const int o = blockIdx.x;   // original output channel 0..127
  const int i = threadIdx.x;  // original input channel 0..127
  const int r = o & 3;        // output residue class
  const int k = o >> 2;       // index within class (og)
  const int op = r * 32 + k;  // reordered output channel o'
  const int ip = (i & 3) * 32 + (i >> 2);  // reordered input channel i'
  const bool m3 = ((i & 3) == r);

  // center (1x1-equivalent) dense weight, K stored in reordered order
  WcR[op * IC + ip] = m3 ? w3[(o * IC + i) * 9 + 4] : w1[o * IC + i];

  if (m3) {
    const int ig = i >> 2;
    int tq = 0;
    for (int tt = 0; tt < 9; ++tt) {
      if (tt == 4) continue;
      WgR[((r * 8 + tq) * 32 + k) * 32 + ig] = w3[(o * IC + i) * 9 + tt];
      ++tq;
    }
  }

  __shared__ float sb[IC];
  sb[i] = bsrc[o * IC + i];
  __syncthreads();
  if (i == 0) {
    float s = 0.f;
    for (int j = 0; j < IC; ++j) s += sb[j];
    biasR[op] = s;
  }
}

// One block: batch b, row h, 32-pixel segment; all 128 output channels.
// 8 waves: wave w -> o' tile [16w, 16w+16), residue group r = w/2.
__global__ __launch_bounds__(256) void hetconv_wmma(
    const float* __restrict__ x, const float* __restrict__ WcR,
    const float* __restrict__ WgR, const float* __restrict__ biasR,
    float* __restrict__ out) {
  // (row, reordered-ch, col) stored at X3[row][col][ch']; ch padded 128->130
  __shared__ float X3[3][NPIX + 2][130];

  const int tid  = threadIdx.x;
  const int wave = tid >> 5;
  const int lane = tid & 31;
  const int n    = lane & 15;  // N index (B/C/D) and M index (A)
  const int kh   = lane >> 4;  // K-half select

  const int tile = blockIdx.x & 3;
  const int h    = (blockIdx.x >> 2) & (HH - 1);
  const int b    = blockIdx.x >> 9;
  const int w0   = tile * NPIX;

  const int opb = wave * 16;        // o' base for this wave
  const int r   = wave >> 1;        // residue group
  const int ogb = (wave & 1) * 16;  // og base within group

  const float* wc_row = WcR + (opb + n) * IC + 2 * kh;
  const float* wg_row0 = WgR + ((r * 8) * 32 + ogb + n) * 32 + 2 * kh;

  // Pull this wave's (L2-resident) weight rows toward the WGP early.
  __builtin_prefetch(wc_row, 0, 0);
  __builtin_prefetch(wg_row0, 0, 0);

  // ---- Phase A: async-DMA halo rows (h-1, h+1) into LDS; zero-fill OOB ----
  {
    const int htotal = 2 * IC * (NPIX + 2);  // 8704 = 34 * 256
    for (int idx = tid; idx < htotal; idx += 256) {
      const int col = idx % (NPIX + 2);
      const int cp  = (idx / (NPIX + 2)) & (IC - 1);      // LDS slot (i')
      const int ch  = 4 * (cp & 31) + (cp >> 5);          // source channel
      const int row = (idx / ((NPIX + 2) * IC)) * 2;      // 0 or 2
      const int hh = h + row - 1;
      const int ww = w0 + col - 1;
      float* lp = &X3[row][col][cp];
      if (hh >= 0 && hh < HH && ww >= 0 && ww < WW) {
        async_load_f32(lp, &x[((b * IC + ch) * HH + hh) * WW + ww]);
      } else {
        *lp = 0.f;
      }
    }
  }

  // ---- Phase B: stage center row h synchronously (needed first) ----
  {
    const int ctotal = IC * (NPIX + 2);  // 4352 = 17 * 256
    for (int idx = tid; idx < ctotal; idx += 256) {
      const int col = idx % (NPIX + 2);
      const int cp  = idx / (NPIX + 2);
      const int ch  = 4 * (cp & 31) + (cp >> 5);
      const int ww = w0 + col - 1;
      float v = 0.f;
      if (ww >= 0 && ww < WW) v = x[((b * IC + ch) * HH + h) * WW + ww];
      X3[1][col][cp] = v;
    }
  }
  __syncthreads();  // row h visible; halo rows still in flight

  const v8f vzero = {0.f, 0.f, 0.f, 0.f, 0.f, 0.f, 0.f, 0.f};
  v8f acc[NT];
  acc[0] = vzero;
  acc[1] = vzero;

  // ---- center pass: dense 1x1, K = 128 (row h only; overlaps halo DMA) ----
  {
    #pragma unroll 4
    for (int kc = 0; kc < IC; kc += 4) {
      v2f a;
      a.x = wc_row[kc];
      a.y = wc_row[kc + 1];
      #pragma unroll
      for (int t = 0; t < NT; ++t) {
        const float* p = &X3[1][1 + 16 * t + n][kc + 2 * kh];
        v2f bm;
        bm.x = p[0];
        bm.y = p[1];
        acc[t] = wmma_f32_16x16x4(a, bm, acc[t]);
      }
    }
  }

  // ---- halo rows must now be resident for the tap pass ----
  wait_asynccnt0();   // my async transfers done
  __syncthreads();    // everyone's async transfers done

  // ---- tap pass: 8 off-center taps, grouped K = 32 (residue r) ----
  // Group-r channels are contiguous LDS slots [32r, 32r+32) -> b64 B loads.
  #pragma unroll
  for (int tq = 0; tq < 8; ++tq) {
    const int tt = (tq < 4) ? tq : tq + 1;  // skip center tap
    const int dy = tt / 3 - 1;
    const int dx = tt % 3 - 1;
    const float* wrow = wg_row0 + (tq * 32) * 32;
    #pragma unroll 2
    for (int kg = 0; kg < 32; kg += 4) {
      v2f a;
      a.x = wrow[kg];
      a.y = wrow[kg + 1];
      const int cb = 32 * r + kg + 2 * kh;  // reordered slot base
      #pragma unroll
      for (int t = 0; t < NT; ++t) {
        const int col = 1 + 16 * t + n + dx;
        const float* p = &X3[1 + dy][col][cb];
        v2f bm;
        bm.x = p[0];
        bm.y = p[1];
        acc[t] = wmma_f32_16x16x4(a, bm, acc[t]);
      }
    }
  }

  // ---- bias + store (C/D layout: VGPR v, lane -> M = v + 8*kh, N = n) ----
  #pragma unroll
  for (int v = 0; v < 8; ++v) {
    const int op = opb + v + 8 * kh;
    const float bv = biasR[op];
    const int o = 4 * (op & 31) + (op >> 5);  // back to original channel
    float* dst = out + ((b * OC + o) * HH + h) * WW + w0 + n;
    #pragma unroll
    for (int t = 0; t < NT; ++t) dst[16 * t] = acc[t][v] + bv;
  }
}

extern "C" void kernel_launch(void* const* d_in, const int* in_sizes, int n_in,
                              void* d_out, int out_size, void* d_ws, size_t ws_size,
                              hipStream_t stream) {
  const float* x  = (const float*)d_in[0];
  const float* w3 = (const float*)d_in[1];
  const float* w1 = (const float*)d_in[2];
  const float* bb = (const float*)d_in[3];
  // d_in[4] is p == 4 (fixed by problem geometry)

  float* WcR   = (float*)d_ws;            // 128*128 floats
  float* WgR   = WcR + IC * OC;           // 4*8*32*32 floats
  float* biasR = WgR + 4 * 8 * 32 * 32;   // 128 floats

  prep_weights<<<OC, IC, 0, stream>>>(w3, w1, bb, WcR, WgR, biasR);
  hetconv_wmma<<<BATCH * HH * (WW / NPIX), 256, 0, stream>>>(
      x, WcR, WgR, biasR, (float*)d_out);
}